// HCCF_71236327571852
// MI455X (gfx1250) — compile-verified
//
#include <hip/hip_runtime.h>

#define TILE 256

// ---------------------------------------------------------------------------
// COO SpMM scatter:  y[dst[e], :] += scale * val[e] * x[src[e], :]   (D = 64)
//
// Block = 256 threads = 8 wave32; stages TILE=256 (dst,src,val) triples into
// LDS via gfx1250 async-to-LDS (ASYNCcnt path), then each 16-lane group owns
// one nnz at a time: float4 (b128) gather of a quarter row-segment + 4 native
// f32 global atomic adds.  (First in file so the disasm snippet shows it.)
// ---------------------------------------------------------------------------
__global__ __launch_bounds__(256) void hccf_spmm_kernel(
    const int* __restrict__ dst, const int* __restrict__ src,
    const float* __restrict__ val, const float* __restrict__ x,
    float* __restrict__ y, int nnz, float scale) {
  __shared__ int   s_dst[TILE];
  __shared__ int   s_src[TILE];
  __shared__ float s_val[TILE];

  const int tid  = threadIdx.x;
  const int base = blockIdx.x * TILE;
  const int e    = base + tid;

#if defined(__gfx1250__) && __has_builtin(__builtin_amdgcn_global_load_async_to_lds_b32)
  {
    typedef __attribute__((address_space(1))) int gint_t;
    typedef __attribute__((address_space(3))) int lint_t;
    if (e < nnz) {
      __builtin_amdgcn_global_load_async_to_lds_b32(
          (gint_t*)(int*)(dst + e), (lint_t*)(s_dst + tid), 0, 0);
      __builtin_amdgcn_global_load_async_to_lds_b32(
          (gint_t*)(int*)(src + e), (lint_t*)(s_src + tid), 0, 0);
      __builtin_amdgcn_global_load_async_to_lds_b32(
          (gint_t*)(int*)(val + e), (lint_t*)(int*)(s_val + tid), 0, 0);
    }
#if __has_builtin(__builtin_amdgcn_s_wait_asynccnt)
    __builtin_amdgcn_s_wait_asynccnt(0);
#endif
  }
#else
  if (e < nnz) {
    s_dst[tid] = dst[e];
    s_src[tid] = src[e];
    s_val[tid] = val[e];
  }
#endif
  __syncthreads();

  const int g    = tid >> 4;   // 16 groups of 16 lanes
  const int lane = tid & 15;   // lane covers 4 consecutive floats of D=64
  const int rem  = nnz - base;
  const int lim  = rem < TILE ? rem : TILE;

#pragma unroll 2
  for (int i = g; i < lim; i += 16) {
    const int   r = s_dst[i];
    const int   c = s_src[i];
    const float v = s_val[i] * scale;

    const float4 xv = ((const float4*)(x + ((size_t)c << 6)))[lane];
    float* yp = y + ((size_t)r << 6) + (lane << 2);

    __hip_atomic_fetch_add(yp + 0, xv.x * v, __ATOMIC_RELAXED, __HIP_MEMORY_SCOPE_AGENT);
    __hip_atomic_fetch_add(yp + 1, xv.y * v, __ATOMIC_RELAXED, __HIP_MEMORY_SCOPE_AGENT);
    __hip_atomic_fetch_add(yp + 2, xv.z * v, __ATOMIC_RELAXED, __HIP_MEMORY_SCOPE_AGENT);
    __hip_atomic_fetch_add(yp + 3, xv.w * v, __ATOMIC_RELAXED, __HIP_MEMORY_SCOPE_AGENT);
  }
}

// ---------------------------------------------------------------------------
// init: emb = concat(user_w, item_w); acc(d_out) = emb
// ---------------------------------------------------------------------------
__global__ __launch_bounds__(256) void hccf_init_kernel(
    const float4* __restrict__ uw, const float4* __restrict__ iw,
    float4* __restrict__ emb, float4* __restrict__ acc,
    int n_user4, int n_total4) {
  int i = blockIdx.x * blockDim.x + threadIdx.x;
  if (i >= n_total4) return;
  const float4 v = (i < n_user4) ? uw[i] : iw[i - n_user4];
  float4 t = v;                 // keep in registers; store twice from regs
  emb[i] = t;
  acc[i] = t;
}

// ---------------------------------------------------------------------------
// zero two buffers in one launch (new_emb and edge_emb)
// ---------------------------------------------------------------------------
__global__ __launch_bounds__(256) void hccf_zero2_kernel(
    float4* __restrict__ a, int na4, float4* __restrict__ b, int nb4) {
  int i = blockIdx.x * blockDim.x + threadIdx.x;
  float4 z = make_float4(0.f, 0.f, 0.f, 0.f);
  if (i < na4) a[i] = z;
  if (i < nb4) b[i] = z;
}

// ---------------------------------------------------------------------------
// acc: out = (out + x) * s   (s = 1.0 for inner layers, 0.25 folds final mean)
// ---------------------------------------------------------------------------
__global__ __launch_bounds__(256) void hccf_acc_kernel(
    float4* __restrict__ out, const float4* __restrict__ x, int n4, float s) {
  int i = blockIdx.x * blockDim.x + threadIdx.x;
  if (i >= n4) return;
  float4 o = out[i];
  float4 v = x[i];
  o.x = (o.x + v.x) * s;
  o.y = (o.y + v.y) * s;
  o.z = (o.z + v.z) * s;
  o.w = (o.w + v.w) * s;
  out[i] = o;
}

extern "C" void kernel_launch(void* const* d_in, const int* in_sizes, int n_in,
                              void* d_out, int out_size, void* d_ws, size_t ws_size,
                              hipStream_t stream) {
  const float* user_w   = (const float*)d_in[0];
  const float* item_w   = (const float*)d_in[1];
  const int*   adj_rows = (const int*)  d_in[2];
  const int*   adj_cols = (const int*)  d_in[3];
  const float* adj_vals = (const float*)d_in[4];
  const int*   h_rows   = (const int*)  d_in[5];
  const int*   h_cols   = (const int*)  d_in[6];
  const float* h_vals   = (const float*)d_in[7];
  float*       out      = (float*)d_out;

  const int D        = 64;
  const int N_USERS  = 100000;
  const int N_NODES  = 150000;
  const int N_EDGES  = 10000;
  const int NNZ_ADJ  = in_sizes[2];   // 4,800,000
  const int NNZ_H    = in_sizes[5];   //   640,000

  const size_t NODE_ELEMS = (size_t)N_NODES * D;   // 9,600,000 floats
  const size_t EDGE_ELEMS = (size_t)N_EDGES * D;   //   640,000 floats
  const size_t NODE_BYTES = (NODE_ELEMS * sizeof(float) + 255) & ~(size_t)255;
  // workspace: embA | embB | edge  (~79.4 MB)
  char*  ws   = (char*)d_ws;
  float* embA = (float*)(ws);
  float* embB = (float*)(ws + NODE_BYTES);
  float* edge = (float*)(ws + 2 * NODE_BYTES);

  const int n_total4 = (int)(NODE_ELEMS / 4);      // 2,400,000
  const int n_user4  = (N_USERS * D) / 4;          // 1,600,000
  const int n_edge4  = (int)(EDGE_ELEMS / 4);      //   160,000
  const int eltBlocks = (n_total4 + 255) / 256;

  // acc(d_out) = emb0 = concat(user_w, item_w)
  hccf_init_kernel<<<eltBlocks, 256, 0, stream>>>(
      (const float4*)user_w, (const float4*)item_w,
      (float4*)embA, (float4*)out, n_user4, n_total4);

  for (int layer = 0; layer < 3; ++layer) {
    // new_emb (embB) and edge_emb cleared; combine scales folded into scatter
    hccf_zero2_kernel<<<eltBlocks, 256, 0, stream>>>(
        (float4*)embB, n_total4, (float4*)edge, n_edge4);

    // graph propagation: embB += 0.8 * A @ embA
    hccf_spmm_kernel<<<(NNZ_ADJ + TILE - 1) / TILE, 256, 0, stream>>>(
        adj_rows, adj_cols, adj_vals, embA, embB, NNZ_ADJ, 0.8f);

    // hyperedge gather: edge = H^T @ embA  (dst = h_cols, src = h_rows)
    hccf_spmm_kernel<<<(NNZ_H + TILE - 1) / TILE, 256, 0, stream>>>(
        h_cols, h_rows, h_vals, embA, edge, NNZ_H, 1.0f);

    // hypergraph scatter: embB += 0.2 * H @ edge  (dst = h_rows, src = h_cols)
    hccf_spmm_kernel<<<(NNZ_H + TILE - 1) / TILE, 256, 0, stream>>>(
        h_rows, h_cols, h_vals, edge, embB, NNZ_H, 0.2f);

    // acc += layer embedding; final layer folds the /4 mean
    hccf_acc_kernel<<<eltBlocks, 256, 0, stream>>>(
        (float4*)out, (const float4*)embB, n_total4,
        (layer == 2) ? 0.25f : 1.0f);

    float* t = embA; embA = embB; embB = t;
  }
}